// TopoRegLoss_49529562857472
// MI455X (gfx1250) — compile-verified
//
#include <hip/hip_runtime.h>
#include <math.h>

// ---------------------------------------------------------------------------
// TopoRegLoss on MI455X (gfx1250):
//   1) row sq-norms                          (trivial, bandwidth)
//   2) D = pairwise L2 via fp32 WMMA GEMM    (v_wmma_f32_16x16x4_f32)
//   3) Prim MST + sum|edge-1|                (single persistent workgroup,
//                                             D resident in 192MB L2)
// ---------------------------------------------------------------------------

#define N_PTS 4096
#define DIM   1024
#define EPS_F 1e-6f
#define TOPS  1.0f
#define INF_F __builtin_inff()

typedef __attribute__((ext_vector_type(2))) float v2f;
typedef __attribute__((ext_vector_type(8))) float v8f;

// ---------------------------------------------------------------- kernel 0 --
// sq[i] = sum_k X[i,k]^2 ; one wave32 per row, float4 loads.
__global__ __launch_bounds__(256) void sqnorm_kernel(const float* __restrict__ X,
                                                     float* __restrict__ sq) {
  const int row  = blockIdx.x * 8 + (threadIdx.x >> 5);
  const int lane = threadIdx.x & 31;
  const float4* xr = (const float4*)(X + (size_t)row * DIM);
  float s = 0.0f;
#pragma unroll
  for (int k = lane; k < DIM / 4; k += 32) {
    float4 v = xr[k];
    s += v.x * v.x + v.y * v.y + v.z * v.z + v.w * v.w;
  }
#pragma unroll
  for (int off = 16; off; off >>= 1) s += __shfl_xor(s, off, 32);
  if (lane == 0) sq[row] = s;
}

// ---------------------------------------------------------------- kernel 1 --
// Each wave computes a 32x32 tile of D using 2x2 fp32 WMMA accumulators.
// A-fragment layout (ISA 7.12.2, 32-bit 16x4): lane L -> M = L%16,
// K = 2*(L/16) + {0,1}. B (4x16) mirrors the K striping with N = L%16.
// C/D layout: VGPR v -> M = v + 8*(L/16), N = L%16.
__global__ __launch_bounds__(256) void dist_wmma_kernel(const float* __restrict__ X,
                                                        const float* __restrict__ sq,
                                                        float* __restrict__ D) {
  const int wave = threadIdx.x >> 5;
  const int lane = threadIdx.x & 31;
  const int tile = blockIdx.x * 8 + wave;        // 128x128 tiles of 32x32
  const int i0 = (tile >> 7) * 32;
  const int j0 = (tile & 127) * 32;

  const int m  = lane & 15;
  const int ko = (lane >> 4) * 2;                // K offset inside the 4-slice

  const float* A0 = X + (size_t)(i0 + m)      * DIM + ko;
  const float* A1 = X + (size_t)(i0 + 16 + m) * DIM + ko;
  const float* B0 = X + (size_t)(j0 + m)      * DIM + ko;
  const float* B1 = X + (size_t)(j0 + 16 + m) * DIM + ko;

  v8f acc00 = {}, acc01 = {}, acc10 = {}, acc11 = {};

  for (int k = 0; k < DIM; k += 4) {
    v2f a0 = *(const v2f*)(A0 + k);
    v2f a1 = *(const v2f*)(A1 + k);
    v2f b0 = *(const v2f*)(B0 + k);
    v2f b1 = *(const v2f*)(B1 + k);
    acc00 = __builtin_amdgcn_wmma_f32_16x16x4_f32(false, a0, false, b0,
                                                  (short)0, acc00, false, false);
    acc01 = __builtin_amdgcn_wmma_f32_16x16x4_f32(false, a0, false, b1,
                                                  (short)0, acc01, false, false);
    acc10 = __builtin_amdgcn_wmma_f32_16x16x4_f32(false, a1, false, b0,
                                                  (short)0, acc10, false, false);
    acc11 = __builtin_amdgcn_wmma_f32_16x16x4_f32(false, a1, false, b1,
                                                  (short)0, acc11, false, false);
  }

  const int n     = lane & 15;
  const int mbase = (lane >> 4) * 8;

  // epilogue: d = sqrt(max(sq_i + sq_j - 2*dot, 0) (+eps where ==0))
#define STORE_TILE(ACC, IB, JB)                                         \
  {                                                                     \
    const float sqj = sq[(JB) + n];                                     \
    _Pragma("unroll")                                                   \
    for (int v = 0; v < 8; ++v) {                                       \
      const int i = (IB) + mbase + v;                                   \
      float d2 = sq[i] + sqj - 2.0f * (ACC)[v];                         \
      d2 = fmaxf(d2, 0.0f);                                             \
      if (d2 == 0.0f) d2 = EPS_F;                                       \
      D[(size_t)i * N_PTS + (JB) + n] = sqrtf(d2);                      \
    }                                                                   \
  }
  STORE_TILE(acc00, i0,      j0)
  STORE_TILE(acc01, i0,      j0 + 16)
  STORE_TILE(acc10, i0 + 16, j0)
  STORE_TILE(acc11, i0 + 16, j0 + 16)
#undef STORE_TILE
}

// ---------------------------------------------------------------- kernel 2 --
// Prim's MST, one persistent 1024-thread (32-wave) workgroup.
// mind[] lives in LDS (16KB); per step: local min over 4 keys, wave32
// shuffle reduction, cross-wave reduction, broadcast j, relax row D[j] (L2).
// Selected weight == mind[j], so loss accumulates directly; no parent array.
__global__ __launch_bounds__(1024) void prim_kernel(const float* __restrict__ D,
                                                    float* __restrict__ out) {
  __shared__ float         mind[N_PTS];
  __shared__ unsigned char visited[N_PTS];
  __shared__ float         wval[32];
  __shared__ int           widx[32];
  __shared__ int           jsel;
  __shared__ float         wsel;

  const int tid  = threadIdx.x;
  const int lane = tid & 31;
  const int wave = tid >> 5;

  for (int t = tid; t < N_PTS; t += 1024) {
    mind[t]    = D[t];              // row 0
    visited[t] = 0;
  }
  __syncthreads();
  if (tid == 0) { visited[0] = 1; mind[0] = INF_F; }
  __syncthreads();

  float loss = 0.0f;

  for (int step = 0; step < N_PTS - 1; ++step) {
    // --- argmin over 4096 keys (lowest index wins ties, as jnp.argmin) ---
    float bv = INF_F; int bi = 0x7fffffff;
#pragma unroll
    for (int q = 0; q < 4; ++q) {
      const int t = tid + q * 1024;
      const float v = mind[t];
      if (v < bv) { bv = v; bi = t; }
    }
#pragma unroll
    for (int off = 16; off; off >>= 1) {
      const float ov = __shfl_xor(bv, off, 32);
      const int   oi = __shfl_xor(bi, off, 32);
      if (ov < bv || (ov == bv && oi < bi)) { bv = ov; bi = oi; }
    }
    if (lane == 0) { wval[wave] = bv; widx[wave] = bi; }
    __syncthreads();
    if (wave == 0) {
      float v = wval[lane]; int idx = widx[lane];
#pragma unroll
      for (int off = 16; off; off >>= 1) {
        const float ov = __shfl_xor(v, off, 32);
        const int   oi = __shfl_xor(idx, off, 32);
        if (ov < v || (ov == v && oi < idx)) { v = ov; idx = oi; }
      }
      if (lane == 0) { jsel = idx; wsel = v; }
    }
    __syncthreads();

    const int j = jsel;
    if (tid == 0) loss += fabsf(wsel - TOPS);

    // --- relax with row D[j]; each LDS slot owned by exactly one thread ---
    const float* Drow = D + (size_t)j * N_PTS;
#pragma unroll
    for (int q = 0; q < 4; ++q) {
      const int t = tid + q * 1024;
      if (t == j) {
        mind[t] = INF_F; visited[t] = 1;
      } else if (!visited[t]) {
        const float dv = Drow[t];
        if (dv < mind[t]) mind[t] = dv;
      }
    }
    __syncthreads();
  }

  if (tid == 0) out[0] = loss;
}

// ------------------------------------------------------------------- launch --
extern "C" void kernel_launch(void* const* d_in, const int* in_sizes, int n_in,
                              void* d_out, int out_size, void* d_ws, size_t ws_size,
                              hipStream_t stream) {
  (void)in_sizes; (void)n_in; (void)out_size; (void)ws_size;
  const float* X = (const float*)d_in[0];

  float* sq = (float*)d_ws;                              // 4096 floats (16KB)
  float* D  = (float*)((char*)d_ws + N_PTS * sizeof(float)); // 4096x4096 f32 (64MB)

  sqnorm_kernel<<<N_PTS / 8, 256, 0, stream>>>(X, sq);
  dist_wmma_kernel<<<(128 * 128) / 8, 256, 0, stream>>>(X, sq, D);
  prim_kernel<<<1, 1024, 0, stream>>>(D, (float*)d_out);
}